// ChamferLoss_2370821948077
// MI455X (gfx1250) — compile-verified
//
#include <hip/hip_runtime.h>

typedef __attribute__((ext_vector_type(2))) float v2f;
typedef __attribute__((ext_vector_type(8))) float v8f;

#define INF_BITS   0x7F800000u
#define COLSPLIT   8          // gridDim.y: column-slice parallelism
#define WAVES_PB   8          // waves per block (256 threads)

__device__ __forceinline__ unsigned umin32(unsigned a, unsigned b) {
    return a < b ? a : b;
}

__global__ void chamfer_init(unsigned int* __restrict__ colMinU,
                             unsigned int* __restrict__ rowMinU, int n, int m) {
    int i = blockIdx.x * blockDim.x + threadIdx.x;
    if (i < m) colMinU[i] = INF_BITS;
    if (i < n) rowMinU[i] = INF_BITS;
}

// Block = 256 threads = 8 waves. Each wave owns one 16-row tile of `pred`
// and sweeps its block's slice of 16-col `label` tiles. The -2*x.y cross
// term is computed with V_WMMA_F32_16X16X4_F32 (K=4, D=3 zero-padded ->
// exact fp32). Min bookkeeping is done in u32 (bit-monotone for d2 >= 0),
// avoiding fminf canonicalization. Labels are software-pipelined 1 deep.
__global__ __launch_bounds__(256) void chamfer_tiles(
    const float* __restrict__ pred, const float* __restrict__ label,
    unsigned int* __restrict__ rowMinU, unsigned int* __restrict__ colMinU,
    int n, int m) {
    __shared__ unsigned int ldsCol[8192 / COLSPLIT];   // 1024 entries = 4 KB

    const int tid  = threadIdx.x;
    const int lane = tid & 31;
    const int wave = tid >> 5;
    const int g    = lane >> 4;                 // 0: K={0,1} half, 1: K={2,3} half
    const int nn   = lane & 15;                 // row (A) / col (B) within tile

    const int numColTiles = m >> 4;
    const int sliceTiles  = numColTiles / COLSPLIT;       // 64
    const int tile0       = blockIdx.y * sliceTiles;      // first col tile of slice
    const int sliceCols   = sliceTiles << 4;              // 1024

    for (int i = tid; i < sliceCols; i += blockDim.x) ldsCol[i] = INF_BITS;
    __syncthreads();

    const int r0 = (blockIdx.x * WAVES_PB + wave) * 16;

    // ---- A fragment: pred rows r0..r0+15, K padded to 4; row norms ----
    v2f a;
    float x2v[8];
    {
        const float* pp = pred + 3 * (r0 + nn);
        float px = pp[0], py = pp[1], pz = pp[2];
        a.x = g ? pz : px;
        a.y = g ? 0.0f : py;
        float x2all = fmaf(px, px, fmaf(py, py, pz * pz)); // lane nn: ||pred[r0+nn]||^2
        #pragma unroll
        for (int j = 0; j < 8; ++j)
            x2v[j] = __shfl(x2all, 8 * g + j);  // ||pred[row of C-vgpr j]||^2
    }

    unsigned rminu[8];
    #pragma unroll
    for (int j = 0; j < 8; ++j) rminu[j] = INF_BITS;

    // ---- tile sweep (rotated start per wave, 1-deep load pipeline) ----
    int ct = wave * (sliceTiles / WAVES_PB);
    {
        const float* lp = label + 3 * (((tile0 + ct) << 4) + nn);
        float cx = lp[0], cy = lp[1], cz = lp[2];

        for (int t = 0; t < sliceTiles; ++t) {
            int ctn = ct + 1; if (ctn == sliceTiles) ctn = 0;
            const float* np = label + 3 * (((tile0 + ctn) << 4) + nn);
            float nx = np[0], ny = np[1], nz = np[2];     // prefetch next tile

            v2f b;
            b.x = g ? cz : cx;                            // cndmask, no branch
            b.y = g ? 0.0f : cy;
            float y2 = fmaf(cx, cx, fmaf(cy, cy, cz * cz));

            v8f acc = {};                                 // C = 0 -> acc = A x B
            acc = __builtin_amdgcn_wmma_f32_16x16x4_f32(
                false, a, false, b, (short)0, acc, false, false);

            unsigned cminu = INF_BITS;
            #pragma unroll
            for (int j = 0; j < 8; ++j) {
                float d2 = fmaf(-2.0f, acc[j], x2v[j] + y2);
                unsigned du = __float_as_uint(d2);        // monotone for d2 >= 0;
                rminu[j] = umin32(rminu[j], du);          // negative noise -> huge uint,
                cminu    = umin32(cminu, du);             // loses min (conservative)
            }
            // Both half-waves min their 8-row partial into the same slot:
            // fire-and-forget DS atomic, no cross-half shuffle, no wait.
            atomicMin(&ldsCol[(ct << 4) + nn], cminu);

            ct = ctn; cx = nx; cy = ny; cz = nz;
        }
    }

    // ---- row mins: u32 min across the 16 lanes of each half, atomic combine ----
    #pragma unroll
    for (int j = 0; j < 8; ++j) {
        unsigned v = rminu[j];
        v = umin32(v, (unsigned)__shfl_xor((int)v, 1));
        v = umin32(v, (unsigned)__shfl_xor((int)v, 2));
        v = umin32(v, (unsigned)__shfl_xor((int)v, 4));
        v = umin32(v, (unsigned)__shfl_xor((int)v, 8));
        rminu[j] = v;
    }
    if (nn == 0) {
        #pragma unroll
        for (int j = 0; j < 8; ++j)
            atomicMin(&rowMinU[r0 + 8 * g + j], rminu[j]);
    }

    // ---- flush block-local column-slice mins to global ----
    __syncthreads();
    const int cBase = tile0 << 4;
    for (int i = tid; i < sliceCols; i += blockDim.x)
        atomicMin(&colMinU[cBase + i], ldsCol[i]);
}

__global__ __launch_bounds__(256) void chamfer_reduce(
    const unsigned int* __restrict__ rowMinU, const unsigned int* __restrict__ colMinU,
    float* __restrict__ out, int n, int m) {
    __shared__ float sdata[256];
    float s = 0.0f;
    const float invN = 1.0f / (float)n, invM = 1.0f / (float)m;
    for (int i = threadIdx.x; i < n; i += 256) s += sqrtf(__uint_as_float(rowMinU[i])) * invN;
    for (int i = threadIdx.x; i < m; i += 256) s += sqrtf(__uint_as_float(colMinU[i])) * invM;
    sdata[threadIdx.x] = s;
    __syncthreads();
    for (int off = 128; off > 0; off >>= 1) {
        if (threadIdx.x < off) sdata[threadIdx.x] += sdata[threadIdx.x + off];
        __syncthreads();
    }
    if (threadIdx.x == 0) out[0] = sdata[0];
}

extern "C" void kernel_launch(void* const* d_in, const int* in_sizes, int n_in,
                              void* d_out, int out_size, void* d_ws, size_t ws_size,
                              hipStream_t stream) {
    const float* pred  = (const float*)d_in[0];
    const float* label = (const float*)d_in[1];
    const int n = in_sizes[0] / 3;    // 8192
    const int m = in_sizes[1] / 3;    // 8192

    unsigned int* rowMinU = (unsigned int*)d_ws;
    unsigned int* colMinU = rowMinU + n;

    const int mx = (n > m) ? n : m;
    chamfer_init<<<(mx + 255) / 256, 256, 0, stream>>>(colMinU, rowMinU, n, m);

    dim3 grid(n / (16 * WAVES_PB), COLSPLIT);          // 64 x 8 = 512 workgroups
    chamfer_tiles<<<grid, 256, 0, stream>>>(pred, label, rowMinU, colMinU, n, m);

    chamfer_reduce<<<1, 256, 0, stream>>>(rowMinU, colMinU, (float*)d_out, n, m);
}